// VectorQuantizer_25159918420456
// MI455X (gfx1250) — compile-verified
//
#include <hip/hip_runtime.h>
#include <hip/hip_bf16.h>
#include <float.h>
#include <math.h>

// VQ-VAE vector quantizer for MI455X (gfx1250, wave32).
// N = 128*64*64 rows, D = 64, K = 128 codes.
// Memory-bound target: ~258 MiB HBM traffic @ 23.3 TB/s ~= 11 us.
// Dot products via native f32 WMMA (V_WMMA_F32_16X16X4_F32): exact f32 MACs,
// matrix-pipe throughput keeps the 32 FLOP/byte intensity under the roofline.

typedef __attribute__((ext_vector_type(2))) float v2f;
typedef __attribute__((ext_vector_type(4))) float v4f;
typedef __attribute__((ext_vector_type(8))) float v8f;

#define K_CODES   128
#define DIM       64
#define ROWS_PB   128                 // rows per block (8 waves x 16 rows)
#define N_ROWS    (128 * 64 * 64)     // 524288
#define ND        (N_ROWS * DIM)      // 33554432
#define LSTRIDE   68                  // LDS row stride (floats): conflict-free b64 frags

__global__ __launch_bounds__(256) void vq_main_kernel(
    const float* __restrict__ x, const float* __restrict__ cb,
    float* __restrict__ out, float* __restrict__ partials,
    unsigned* __restrict__ hist)
{
    __shared__ float ldsE[K_CODES * LSTRIDE];   // codebook, padded rows
    __shared__ float ldsX[ROWS_PB * LSTRIDE];   // input tile, padded rows
    __shared__ float ldsEn[K_CODES];            // ||e||^2
    __shared__ int   ldsIdx[ROWS_PB];           // winning code per row
    __shared__ float ldsRed[256];

    const int tid = threadIdx.x;
    const long long rowBase = (long long)blockIdx.x * ROWS_PB;

    // ---- Stage codebook (32 KB) into LDS: coalesced b128 global loads ----
    for (int g = tid * 4; g < K_CODES * DIM; g += 256 * 4) {
        v4f v = *(const v4f*)(cb + g);
        int r = g >> 6, c = g & 63;
        ldsE[r * LSTRIDE + c + 0] = v.x;
        ldsE[r * LSTRIDE + c + 1] = v.y;
        ldsE[r * LSTRIDE + c + 2] = v.z;
        ldsE[r * LSTRIDE + c + 3] = v.w;
    }
    // ---- Stage x tile (32 KB) into LDS: coalesced b128 global loads ----
    const float* xin = x + rowBase * DIM;
    for (int g = tid * 4; g < ROWS_PB * DIM; g += 256 * 4) {
        v4f v = *(const v4f*)(xin + g);
        int r = g >> 6, c = g & 63;
        ldsX[r * LSTRIDE + c + 0] = v.x;
        ldsX[r * LSTRIDE + c + 1] = v.y;
        ldsX[r * LSTRIDE + c + 2] = v.z;
        ldsX[r * LSTRIDE + c + 3] = v.w;
    }
    __syncthreads();

    if (tid < K_CODES) {
        float s = 0.f;
        #pragma unroll 8
        for (int c = 0; c < DIM; ++c) { float e = ldsE[tid * LSTRIDE + c]; s += e * e; }
        ldsEn[tid] = s;
    }
    __syncthreads();

    // ---- WMMA: dot(x_tile16, E_tile16^T) over D=64 in K=4 chunks ----
    const int wave  = tid >> 5;
    const int lane  = tid & 31;
    const int row16 = lane & 15;     // M row (A) and N col (B) within the tile
    const int half  = lane >> 4;     // selects K offset {0,2} within chunk
    const int mBase = wave * 16;

    v8f acc[8];
    #pragma unroll
    for (int ct = 0; ct < 8; ++ct) acc[ct] = {};

    const float* aRow = &ldsX[(mBase + row16) * LSTRIDE];
    #pragma unroll
    for (int kk = 0; kk < 16; ++kk) {
        const int kb = kk * 4 + half * 2;
        v2f a = *(const v2f*)(aRow + kb);                 // A frag: ds_load_b64
        #pragma unroll
        for (int ct = 0; ct < 8; ++ct) {
            v2f b = *(const v2f*)(&ldsE[(ct * 16 + row16) * LSTRIDE + kb]);
            acc[ct] = __builtin_amdgcn_wmma_f32_16x16x4_f32(
                false, a, false, b, (short)0, acc[ct], false, false);
        }
    }

    // ---- distances -> per-lane running argmin over 8 code tiles ----
    float best[8]; int bidx[8];
    #pragma unroll
    for (int j = 0; j < 8; ++j) { best[j] = FLT_MAX; bidx[j] = 0x7fffffff; }

    #pragma unroll
    for (int ct = 0; ct < 8; ++ct) {
        const float en  = ldsEn[ct * 16 + row16];
        const int   idx = ct * 16 + row16;
        #pragma unroll
        for (int j = 0; j < 8; ++j) {
            float s = en - 2.0f * acc[ct][j];   // ||x||^2 constant per row: drop it
            if (s < best[j]) { best[j] = s; bidx[j] = idx; }
        }
    }

    // ---- min-reduce across the 16-lane N dimension (xor 1,2,4,8) ----
    #pragma unroll
    for (int j = 0; j < 8; ++j) {
        #pragma unroll
        for (int off = 1; off < 16; off <<= 1) {
            float ob = __shfl_xor(best[j], off, 32);
            int   oi = __shfl_xor(bidx[j], off, 32);
            if (ob < best[j] || (ob == best[j] && oi < bidx[j])) {
                best[j] = ob; bidx[j] = oi;
            }
        }
    }
    // lane (row16==0) of each half holds rows mBase + half*8 + j
    if (row16 == 0) {
        #pragma unroll
        for (int j = 0; j < 8; ++j) ldsIdx[mBase + half * 8 + j] = bidx[j];
    }
    __syncthreads();

    // ---- Gather quantized rows, write coalesced, accumulate (q-x)^2 ----
    float sum = 0.f;
    float* qout = out + 1 + rowBase * DIM;
    #pragma unroll 4
    for (int k = 0; k < 32; ++k) {
        int g = k * 256 + tid;        // flat index in the 128x64 tile
        int r = g >> 6, c = g & 63;
        int idx = ldsIdx[r];
        float q  = ldsE[idx * LSTRIDE + c];
        float xv = ldsX[r * LSTRIDE + c];
        float d = q - xv;
        sum += d * d;
        qout[g] = q;                  // coalesced global_store_b32
    }

    // deterministic per-block reduction of squared error
    ldsRed[tid] = sum;
    __syncthreads();
    #pragma unroll
    for (int s = 128; s > 0; s >>= 1) {
        if (tid < s) ldsRed[tid] += ldsRed[tid + s];
        __syncthreads();
    }
    if (tid == 0) partials[blockIdx.x] = ldsRed[0];

    // histogram (integer atomics: deterministic) + index output (as float)
    if (tid < ROWS_PB) {
        atomicAdd(&hist[ldsIdx[tid] & (K_CODES - 1)], 1u);
        out[2 + (long long)ND + rowBase + tid] = (float)ldsIdx[tid];
    }
}

__global__ __launch_bounds__(256) void vq_finalize_kernel(
    const float* __restrict__ partials, int nPart,
    const unsigned* __restrict__ hist, float* __restrict__ out)
{
    __shared__ float red[256];
    const int tid = threadIdx.x;

    // loss = (1 + 0.25) * mean((q - x)^2)  (q-loss == e-loss numerically)
    float s = 0.f;
    for (int i = tid; i < nPart; i += 256) s += partials[i];
    red[tid] = s;
    __syncthreads();
    #pragma unroll
    for (int st = 128; st > 0; st >>= 1) {
        if (tid < st) red[tid] += red[tid + st];
        __syncthreads();
    }
    if (tid == 0) out[0] = 1.25f * red[0] / (float)ND;
    __syncthreads();

    // perplexity = exp(-sum p log(p + 1e-10))
    float h = 0.f;
    if (tid < K_CODES) {
        float p = (float)hist[tid] / (float)N_ROWS;
        h = p * logf(p + 1e-10f);
    }
    red[tid] = h;
    __syncthreads();
    #pragma unroll
    for (int st = 128; st > 0; st >>= 1) {
        if (tid < st) red[tid] += red[tid + st];
        __syncthreads();
    }
    if (tid == 0) out[1 + (long long)ND] = expf(-red[0]);
}

extern "C" void kernel_launch(void* const* d_in, const int* in_sizes, int n_in,
                              void* d_out, int out_size, void* d_ws, size_t ws_size,
                              hipStream_t stream) {
    const float* x  = (const float*)d_in[0];   // [128,64,64,64] f32
    const float* cb = (const float*)d_in[1];   // [128,64] f32
    float* out = (float*)d_out;                // [loss | quantized(ND) | perplexity | indices(N)]

    unsigned* hist    = (unsigned*)d_ws;               // 128 u32
    float*    partials = (float*)d_ws + K_CODES;       // 4096 f32

    const int nBlocks = N_ROWS / ROWS_PB;              // 4096

    hipMemsetAsync(d_ws, 0, K_CODES * sizeof(unsigned), stream);
    vq_main_kernel<<<nBlocks, 256, 0, stream>>>(x, cb, out, partials, hist);
    vq_finalize_kernel<<<1, 256, 0, stream>>>(partials, nBlocks, hist, out);
}